// ChaCha20Cipher_30580167148128
// MI455X (gfx1250) — compile-verified
//
#include <hip/hip_runtime.h>
#include <cstdint>

#define TPB 256

typedef float v2f __attribute__((ext_vector_type(2)));

// per-lane tanh (V_TANH_F32 is scalar-only; confirmed selected in round-1/2 codegen)
__device__ __forceinline__ v2f tanh2(v2f t) {
#if __has_builtin(__builtin_amdgcn_tanhf)
  return v2f{__builtin_amdgcn_tanhf(t.x), __builtin_amdgcn_tanhf(t.y)};
#else
  float ex = __builtin_amdgcn_exp2f(2.8853900817779268f * t.x); // e^(2t)
  float ey = __builtin_amdgcn_exp2f(2.8853900817779268f * t.y);
  return v2f{1.0f - 2.0f * __builtin_amdgcn_rcpf(ex + 1.0f),
             1.0f - 2.0f * __builtin_amdgcn_rcpf(ey + 1.0f)};
#endif
}

__device__ __forceinline__ v2f v2floor(v2f x) {
  return v2f{__builtin_floorf(x.x), __builtin_floorf(x.y)};
}

// soft_add(x,y) = s - sigmoid(10(s-1)) = s - 0.5 - 0.5*tanh(5s-5)
__device__ __forceinline__ v2f soft_add(v2f x, v2f y) {
  v2f s = x + y;
  v2f h = tanh2(s * 5.0f - 5.0f);          // pk_fma
  return (s - 0.5f) - 0.5f * h;            // pk_add + pk_fma
}

// soft_xor via h = tanh(5x-2.5):  t1+t2 = (1-hx*hy)/2,  t1*t2 = (1-hx^2)(1-hy^2)/16
__device__ __forceinline__ v2f soft_xor(v2f x, v2f y) {
  v2f hx = tanh2(x * 5.0f - 2.5f);         // pk_fma + 2x tanh
  v2f hy = tanh2(y * 5.0f - 2.5f);
  v2f first = 0.5f - 0.5f * (hx * hy);     // pk_mul + pk_fma
  v2f a = 1.0f - hx * hx;                  // pk_fma
  v2f b = 1.0f - hy * hy;                  // pk_fma
  v2f r = first - 0.0625f * (a * b);       // pk_mul + pk_fma
  r.x = __builtin_amdgcn_fmed3f(r.x, 0.0f, 1.0f);
  r.y = __builtin_amdgcn_fmed3f(r.y, 0.0f, 1.0f);
  return r;
}

// rotate_left: all scales are exact powers of two, so the reference's fp32 double-fmod
// chain reduces exactly to: y=x*2^N; t = fl(fract(y) + y*2^-32); result = fract(t)
template <int N>
__device__ __forceinline__ v2f rotl(v2f x) {
  const float RMODF = 2.3283064365386963e-10f; // 2^-32 (exact)
  v2f y = x * (float)(1u << N);              // pk_mul (exact)
  v2f f = y - v2floor(y);                    // 2x v_floor + pk_add
  v2f t = f + y * RMODF;                     // pk_fma (the one rounded step)
  return t - v2floor(t);                     // 2x v_floor + pk_add
}

__device__ __forceinline__ void quarter_round(v2f& a, v2f& b, v2f& c, v2f& d) {
  a = soft_add(a, b); d = soft_xor(d, a); d = rotl<16>(d);
  c = soft_add(c, d); b = soft_xor(b, c); b = rotl<12>(b);
  a = soft_add(a, b); d = soft_xor(d, a); d = rotl<8>(d);
  c = soft_add(c, d); b = soft_xor(b, c); b = rotl<7>(b);
}

__global__ __launch_bounds__(TPB) void chacha20_soft_kernel(
    const float* __restrict__ pt, const float* __restrict__ key,
    const float* __restrict__ nonce, float* __restrict__ out, int npairs) {
  // 2 batch elements per thread -> 32 plaintext floats/thread; 33-float row pitch
  // ((lane*33) % 64 banks injective across a wave -> conflict-free LDS)
  __shared__ float tile[TPB * 33];
  const int lt = threadIdx.x;
  const long long gid = (long long)blockIdx.x * TPB + lt; // pair index
  if (gid >= npairs) return;

  // ---- Async-stage 128 B of plaintext (2 rows) global->LDS; consumed only at the end,
  // hidden behind ~2000 TRANS ops of compute, and keeps 32 floats/thread out of VGPRs.
  unsigned ldsOff = (unsigned)(uintptr_t)&tile[lt * 33];
  unsigned vOff   = (unsigned)(gid * 128ll);
  asm volatile(
      "global_load_async_to_lds_b128 %0, %1, %2 offset:0\n\t"
      "global_load_async_to_lds_b128 %0, %1, %2 offset:16\n\t"
      "global_load_async_to_lds_b128 %0, %1, %2 offset:32\n\t"
      "global_load_async_to_lds_b128 %0, %1, %2 offset:48\n\t"
      "global_load_async_to_lds_b128 %0, %1, %2 offset:64\n\t"
      "global_load_async_to_lds_b128 %0, %1, %2 offset:80\n\t"
      "global_load_async_to_lds_b128 %0, %1, %2 offset:96\n\t"
      "global_load_async_to_lds_b128 %0, %1, %2 offset:112"
      :
      : "v"(ldsOff), "v"(vOff), "s"(pt)
      : "memory");

  // ---- keys for both elements (4x b128) and nonces (3x b64; 24 B, 8-byte aligned)
  const float4* kp = (const float4*)(key + gid * 16);
  const float4 ka0 = kp[0], ka1 = kp[1], kb0 = kp[2], kb1 = kp[3];
  const float2* np = (const float2*)(nonce + gid * 6);
  const float2 nn0 = np[0], nn1 = np[1], nn2 = np[2];
  // element A nonce: nn0.x, nn0.y, nn1.x ; element B nonce: nn1.y, nn2.x, nn2.y

  const float C0 = (float)(1634760805.0 / 4294967296.0);
  const float C1 = (float)(857760878.0 / 4294967296.0);
  const float C2 = (float)(2036477234.0 / 4294967296.0);
  const float C3 = (float)(1797285236.0 / 4294967296.0);

  v2f w0 = v2f{C0, C0}, w1 = v2f{C1, C1}, w2 = v2f{C2, C2}, w3 = v2f{C3, C3};
  v2f w4 = v2f{ka0.x, kb0.x}, w5  = v2f{ka0.y, kb0.y};
  v2f w6 = v2f{ka0.z, kb0.z}, w7  = v2f{ka0.w, kb0.w};
  v2f w8 = v2f{ka1.x, kb1.x}, w9  = v2f{ka1.y, kb1.y};
  v2f w10 = v2f{ka1.z, kb1.z}, w11 = v2f{ka1.w, kb1.w};
  v2f w12 = v2f{0.0f, 0.0f};
  v2f w13 = v2f{nn0.x, nn1.y};
  v2f w14 = v2f{nn0.y, nn2.x};
  v2f w15 = v2f{nn1.x, nn2.y};

  const v2f i4 = w4, i5 = w5, i6 = w6, i7 = w7, i8 = w8, i9 = w9, i10 = w10, i11 = w11;
  const v2f i13 = w13, i14 = w14, i15 = w15;

#pragma unroll 1
  for (int r = 0; r < 10; ++r) {
    quarter_round(w0, w4, w8,  w12);
    quarter_round(w1, w5, w9,  w13);
    quarter_round(w2, w6, w10, w14);
    quarter_round(w3, w7, w11, w15);
    quarter_round(w0, w5, w10, w15);
    quarter_round(w1, w6, w11, w12);
    quarter_round(w2, w7, w8,  w13);
    quarter_round(w3, w4, w9,  w14);
  }

  // keystream = soft_add(w, initial_state)
  w0  = soft_add(w0,  v2f{C0, C0}); w1  = soft_add(w1,  v2f{C1, C1});
  w2  = soft_add(w2,  v2f{C2, C2}); w3  = soft_add(w3,  v2f{C3, C3});
  w4  = soft_add(w4,  i4);  w5  = soft_add(w5,  i5);
  w6  = soft_add(w6,  i6);  w7  = soft_add(w7,  i7);
  w8  = soft_add(w8,  i8);  w9  = soft_add(w9,  i9);
  w10 = soft_add(w10, i10); w11 = soft_add(w11, i11);
  w12 = soft_add(w12, v2f{0.0f, 0.0f});
  w13 = soft_add(w13, i13); w14 = soft_add(w14, i14); w15 = soft_add(w15, i15);

  // ---- plaintext has long since landed in LDS
  asm volatile("s_wait_asynccnt 0" ::: "memory");
  const float* tp = &tile[lt * 33];

  v2f res[16];
  {
    v2f ww[16] = {w0, w1, w2, w3, w4, w5, w6, w7, w8, w9, w10, w11, w12, w13, w14, w15};
#pragma unroll
    for (int i = 0; i < 16; ++i)
      res[i] = soft_xor(v2f{tp[i], tp[16 + i]}, ww[i]);
  }

  float4* o = (float4*)(out + gid * 32);
  o[0] = make_float4(res[0].x,  res[1].x,  res[2].x,  res[3].x);
  o[1] = make_float4(res[4].x,  res[5].x,  res[6].x,  res[7].x);
  o[2] = make_float4(res[8].x,  res[9].x,  res[10].x, res[11].x);
  o[3] = make_float4(res[12].x, res[13].x, res[14].x, res[15].x);
  o[4] = make_float4(res[0].y,  res[1].y,  res[2].y,  res[3].y);
  o[5] = make_float4(res[4].y,  res[5].y,  res[6].y,  res[7].y);
  o[6] = make_float4(res[8].y,  res[9].y,  res[10].y, res[11].y);
  o[7] = make_float4(res[12].y, res[13].y, res[14].y, res[15].y);
}

extern "C" void kernel_launch(void* const* d_in, const int* in_sizes, int n_in,
                              void* d_out, int out_size, void* d_ws, size_t ws_size,
                              hipStream_t stream) {
  const float* pt    = (const float*)d_in[0];
  const float* key   = (const float*)d_in[1];
  const float* nonce = (const float*)d_in[2];
  float* out = (float*)d_out;

  const int n = in_sizes[0] / 16;      // batch size B (even: 2^21)
  const int npairs = n / 2;            // 2 elements per thread
  const int grid = (npairs + TPB - 1) / TPB;
  chacha20_soft_kernel<<<grid, TPB, 0, stream>>>(pt, key, nonce, out, npairs);
}